// PillarFeatureNet_25881472926249
// MI455X (gfx1250) — compile-verified
//
#include <hip/hip_runtime.h>
#include <hip/hip_bf16.h>

// PillarFeatureNet scatter-sum: out[c, bin] += x[p, c] for bin = indices[p].
// out layout: (C=64, 512*512) flattened; bin in [0, 262144).

#define N_PILLARS (512 * 512)
#define N_FEAT 64
#define WAVES_PER_BLOCK 8
#define BLOCK_THREADS (WAVES_PER_BLOCK * 32)

// ---------------------------------------------------------------------------
// gfx1250 async copy: each lane moves 16 bytes global -> LDS; one wave moves
// 512 B = TWO points' feature vectors (2 x 64 f32) per issue, fully coalesced.
// Tracked with ASYNCcnt (one increment per instruction for the whole wave).
// ---------------------------------------------------------------------------
__device__ __forceinline__ void async_copy_pair_b128(unsigned lds_byte_off,
                                                     const float* src) {
  unsigned long long ga = (unsigned long long)(uintptr_t)src;
  asm volatile("global_load_async_to_lds_b128 %0, %1, off"
               :
               : "v"(lds_byte_off), "v"(ga)
               : "memory");
}

// ---------------------------------------------------------------------------
// Scatter-add kernel (defined FIRST so the disasm snippet shows it).
// One wave owns a contiguous pair of points per pipeline stage, grid-strided
// over pairs so active waves stream adjacent 512 B slices of x. Double-
// buffered async-to-LDS pipeline hides load latency behind the atomic stream.
// Output RMW stays resident in the 192 MB L2 (output = 64 MB).
// ---------------------------------------------------------------------------
__global__ __launch_bounds__(BLOCK_THREADS) void pfn_scatter_kernel(
    const float* __restrict__ x, const int* __restrict__ indices,
    float* __restrict__ out, int n_points) {
  // 8 waves x 2 buffers x 128 floats (2 points) = 8 KB LDS.
  __shared__ float stage[WAVES_PER_BLOCK][2][2 * N_FEAT];

  const int lane = threadIdx.x & 31;
  const int wave = threadIdx.x >> 5;
  const int gwave = blockIdx.x * WAVES_PER_BLOCK + wave;
  const int nwave = gridDim.x * WAVES_PER_BLOCK;

  // Wave-relative LDS byte offsets of this wave's two staging buffers;
  // lane l owns bytes [16*l, 16*l+16) of each 512 B buffer.
  const unsigned ldsA =
      (unsigned)(uintptr_t)(&stage[wave][0][0]) + (unsigned)lane * 16u;
  const unsigned ldsB =
      (unsigned)(uintptr_t)(&stage[wave][1][0]) + (unsigned)lane * 16u;

  const int n_pairs = n_points >> 1;

  int q = gwave;
  if (q < n_pairs) {
    async_copy_pair_b128(ldsA, x + (size_t)q * (2 * N_FEAT));  // prime
  }

  int sel = 0;  // buffer currently being consumed
  for (; q < n_pairs; q += nwave) {
    const int qn = q + nwave;
    const bool pf = (qn < n_pairs);
    if (pf) {
      async_copy_pair_b128(sel ? ldsA : ldsB, x + (size_t)qn * (2 * N_FEAT));
    }

    // Two consecutive bins -> one b64 load (same address in all lanes).
    const int2 bins = *(const int2*)(indices + 2 * (size_t)q);

    // Retire the load feeding the current buffer (async loads complete in
    // order; allow the prefetch to remain outstanding).
    if (pf) {
      asm volatile("s_wait_asynccnt 0x1" ::: "memory");
    } else {
      asm volatile("s_wait_asynccnt 0x0" ::: "memory");
    }

    // Lane l consumes features l and l+32 of both points: banks 0..31 and
    // 32..63, conflict-free.
    const float a0 = stage[wave][sel][lane];
    const float a1 = stage[wave][sel][lane + 32];
    const float b0 = stage[wave][sel][lane + 64];
    const float b1 = stage[wave][sel][lane + 96];

    // out is (C, 262144): feature c scatters to out[c * N_PILLARS + bin].
    float* baseA = out + bins.x;
    float* baseB = out + bins.y;
    (void)unsafeAtomicAdd(baseA + (size_t)lane * N_PILLARS, a0);
    (void)unsafeAtomicAdd(baseA + (size_t)(lane + 32) * N_PILLARS, a1);
    (void)unsafeAtomicAdd(baseB + (size_t)lane * N_PILLARS, b0);
    (void)unsafeAtomicAdd(baseB + (size_t)(lane + 32) * N_PILLARS, b1);

    sel ^= 1;
  }

  // Odd-point tail (n_points = 2M is even; kept for generality): direct path.
  if ((n_points & 1) && gwave == 0) {
    const int p = n_points - 1;
    const float v0 = x[(size_t)p * N_FEAT + lane];
    const float v1 = x[(size_t)p * N_FEAT + lane + 32];
    const int bin = indices[p];
    float* base = out + bin;
    (void)unsafeAtomicAdd(base + (size_t)lane * N_PILLARS, v0);
    (void)unsafeAtomicAdd(base + (size_t)(lane + 32) * N_PILLARS, v1);
  }
}

// ---------------------------------------------------------------------------
// Zero-fill output (harness poisons d_out with 0xAA before timing).
// float4 stores; 64 MB -> lands in (and warms) the 192 MB L2.
// ---------------------------------------------------------------------------
__global__ __launch_bounds__(256) void pfn_zero_kernel(float4* __restrict__ out,
                                                       int n4) {
  int i = blockIdx.x * blockDim.x + threadIdx.x;
  if (i < n4) out[i] = make_float4(0.f, 0.f, 0.f, 0.f);
}

// ---------------------------------------------------------------------------
// Launch
// ---------------------------------------------------------------------------
extern "C" void kernel_launch(void* const* d_in, const int* in_sizes, int n_in,
                              void* d_out, int out_size, void* d_ws, size_t ws_size,
                              hipStream_t stream) {
  const float* x = (const float*)d_in[0];      // (N_POINTS, 64) f32
  const int* indices = (const int*)d_in[1];    // (N_POINTS,) integer bins
  float* out = (float*)d_out;                  // (64, 512, 512) f32
  const int n_points = in_sizes[1];

  const int n4 = out_size / 4;
  pfn_zero_kernel<<<(n4 + 255) / 256, 256, 0, stream>>>((float4*)out, n4);

  // 2048 blocks * 8 waves = 16384 waves; ~61 grid-strided pairs per wave.
  const int blocks = 2048;
  pfn_scatter_kernel<<<blocks, BLOCK_THREADS, 0, stream>>>(x, indices, out,
                                                           n_points);
}